// SIFLayerMask_50835232916011
// MI455X (gfx1250) — compile-verified
//
#include <hip/hip_runtime.h>
#include <hip/hip_bf16.h>

// ---------------------------------------------------------------------------
// SIFLayerMask on gfx1250:
//   1) polar grid-sample (VALU)
//   2) 15x15 wrap conv: async-to-LDS staging + V_WMMA_F32_16X16X4_F32
//   3) Gram feats@feats.T: b128-fed V_WMMA_F32_16X16X4_F32, K split over 4 waves
// ---------------------------------------------------------------------------

#define BB   64
#define HH   480
#define WW   640
#define PHH  64
#define PWW  512
#define FS   15
#define NF   7
#define RR   7          // FS/2
#define KPAD 228        // 225 padded up to multiple of 4

#define CODES_SIZE (BB * NF * PHH * PWW)   // 14,680,064
#define GRAM_DIM   (BB * NF)               // 448
#define GRAM_SIZE  (GRAM_DIM * GRAM_DIM)   // 200,704
#define POLAR_SIZE (BB * PHH * PWW)        // 2,097,152

#define HALO_W   78                         // 64 + 14
#define STAGE_W  80                         // padded LDS row stride
#define STAGE_N  (FS * HALO_W)              // 1170 staged elements

typedef __attribute__((ext_vector_type(2))) float v2f;
typedef __attribute__((ext_vector_type(4))) float v4f;
typedef __attribute__((ext_vector_type(8))) float v8f;

// ---------------------------------------------------------------------------
// Kernel 1: polar bilinear sampling (grid_sample, align_corners=False, zeros)
// ---------------------------------------------------------------------------
__global__ __launch_bounds__(256) void sif_polar_kernel(
    const float* __restrict__ img,   // [B,1,H,W]
    const float* __restrict__ pup,   // [B,3]
    const float* __restrict__ iri,   // [B,3]
    float* __restrict__ polar)       // [B,PH,PW]
{
    int idx = blockIdx.x * blockDim.x + threadIdx.x;
    if (idx >= POLAR_SIZE) return;
    int j = idx & (PWW - 1);
    int i = (idx >> 9) & (PHH - 1);
    int b = idx >> 15;

    float px = pup[b * 3 + 0], py = pup[b * 3 + 1], pr = pup[b * 3 + 2];
    float cx = iri[b * 3 + 0], cy = iri[b * 3 + 1], cr = iri[b * 3 + 2];

    float theta = 6.283185307179586f * (float)(j + 1) / (float)PWW;
    float ct = cosf(theta), st = sinf(theta);

    float pxc = px + pr * ct, pyc = py + pr * st;
    float ixc = cx + cr * ct, iyc = cy + cr * st;

    float rad = (float)i / (float)(PHH - 1);
    float xg = (1.0f - rad) * pxc + rad * ixc;
    float yg = (1.0f - rad) * pyc + rad * iyc;

    float x = fminf(fmaxf(xg, 0.0f), (float)(WW - 1));
    float y = fminf(fmaxf(yg, 0.0f), (float)(HH - 1));
    float xn = x / (float)(WW - 1) * 2.0f - 1.0f;
    float yn = y / (float)(HH - 1) * 2.0f - 1.0f;
    float gx = ((xn + 1.0f) * 0.5f * (float)WW - 0.5f) / (float)(WW - 1) * 2.0f - 1.0f;
    float gy = ((yn + 1.0f) * 0.5f * (float)HH - 0.5f) / (float)(HH - 1) * 2.0f - 1.0f;

    float xp = ((gx + 1.0f) * (float)WW - 1.0f) * 0.5f;
    float yp = ((gy + 1.0f) * (float)HH - 1.0f) * 0.5f;

    float x0 = floorf(xp), y0 = floorf(yp);
    float wx = xp - x0, wy = yp - y0;

    const float* im = img + (size_t)b * (HH * WW);
    auto gather = [&](float fx, float fy) -> float {
        bool valid = (fx >= 0.0f) && (fx < (float)WW) && (fy >= 0.0f) && (fy < (float)HH);
        int xi = (int)fminf(fmaxf(fx, 0.0f), (float)(WW - 1));
        int yi = (int)fminf(fmaxf(fy, 0.0f), (float)(HH - 1));
        float v = im[yi * WW + xi];
        return valid ? v : 0.0f;
    };
    float v00 = gather(x0,        y0);
    float v01 = gather(x0 + 1.0f, y0);
    float v10 = gather(x0,        y0 + 1.0f);
    float v11 = gather(x0 + 1.0f, y0 + 1.0f);

    float out = v00 * (1.0f - wx) * (1.0f - wy) + v01 * wx * (1.0f - wy) +
                v10 * (1.0f - wx) * wy          + v11 * wx * wy;
    polar[idx] = out * 255.0f;
}

// ---------------------------------------------------------------------------
// Kernel 2: 15x15 wrap conv. Block = 4 waves = (b, h, 64-wide w segment).
// Wrapped 15x78 polar halo staged into LDS via GLOBAL_LOAD_ASYNC_TO_LDS_B32
// (ASYNCcnt), then im2col WMMA f32 16x16x4 fed entirely from LDS.
// ---------------------------------------------------------------------------
__global__ __launch_bounds__(128) void sif_conv_wmma_kernel(
    const float* __restrict__ polar,  // [B,PH,PW]
    const float* __restrict__ filt,   // [FS,FS,NF]
    float* __restrict__ codes)        // [B,NF,PH,PW]
{
    __shared__ __align__(16) float stage[FS * STAGE_W];   // 4800 B
    __shared__ __align__(16) float B2[16 * KPAD];         // 14592 B, [n][k]
    __shared__ __align__(16) int   tab[KPAD];             // 912 B, byte offsets

    int tid = threadIdx.x;
    // block -> (b, h, w0)
    int wseg = blockIdx.x & 7;            // 8 segments of 64 along w
    int h    = (blockIdx.x >> 3) & 63;
    int b    = blockIdx.x >> 9;
    int w0   = wseg << 6;

    const float* pb = polar + ((size_t)b << 15);   // b * PH * PW

    // --- async staging of wrapped halo tile: rows h-7..h+7, cols w0-7..w0+70
    for (int e = tid; e < STAGE_N; e += 128) {
        int kh = e / HALO_W;
        int c  = e - kh * HALO_W;
        int hh = (h + kh + (PHH - RR)) & (PHH - 1);
        int wg = (w0 + c + (PWW - RR)) & (PWW - 1);
        unsigned lds_off = (unsigned)(uintptr_t)(stage + (kh * STAGE_W + c));
        unsigned long long ga = (unsigned long long)(uintptr_t)(pb + ((hh << 9) + wg));
        asm volatile("global_load_async_to_lds_b32 %0, %1, off"
                     :: "v"(lds_off), "v"(ga) : "memory");
    }

    // --- filters (channel-flipped, zero padded) transposed [n][k] in LDS
    for (int t = tid; t < 16 * KPAD; t += 128) {
        int n = t / KPAD;
        int k = t - n * KPAD;
        float v = (k < FS * FS && n < NF) ? filt[k * NF + (NF - 1 - n)] : 0.0f;
        B2[t] = v;
    }
    // --- im2col byte-offset table
    for (int k = tid; k < KPAD; k += 128) {
        int kk = (k < FS * FS) ? k : 0;   // padded taps hit B==0 anyway
        tab[k] = ((kk / FS) * STAGE_W + (kk % FS)) * 4;
    }

    asm volatile("s_wait_asynccnt 0x0" ::: "memory");  // own async copies done
    __syncthreads();                                    // all waves' copies done

    int wave = tid >> 5, lane = tid & 31;
    int m    = lane & 15;                  // A-row (pixel) / D-col (filter)
    int koff = (lane < 16) ? 0 : 2;        // instruction-local K pair per lane half
    int pb4  = (wave * 16 + m) * 4;        // pixel column base, bytes

    const char* stg = (const char*)stage;
    const float* brow = &B2[m * KPAD];

    v8f acc = {};
    #pragma unroll 3
    for (int k0 = 0; k0 < KPAD; k0 += 4) {   // 57 iterations
        int ka = k0 + koff;
        int2 tp = *(const int2*)&tab[ka];    // ds_load_b64: {tab[ka], tab[ka+1]}
        v2f a;
        a.x = *(const float*)(stg + tp.x + pb4);
        a.y = *(const float*)(stg + tp.y + pb4);
        v2f bv = *(const v2f*)(brow + ka);   // ds_load_b64
        acc = __builtin_amdgcn_wmma_f32_16x16x4_f32(
            false, a, false, bv, (short)0, acc, false, false);
    }

    // D: VGPR v -> M = v + (lane<16?0:8) = w within tile, N = lane%16 = filter
    if (m < NF) {
        size_t base = ((((size_t)(b * NF + m)) * PHH + h) << 9) +
                      (size_t)(w0 + wave * 16 + ((lane < 16) ? 0 : 8));
        *(v8f*)(codes + base) = acc;  // 32B-aligned, 8 consecutive w
    }
}

// ---------------------------------------------------------------------------
// Kernel 3: Gram = feats @ feats.T / 14680064, feats = [448][32768].
// One block per 16x16 tile; 4 waves split K (8192 each), LDS reduce.
// b128 loads: lane<16 takes k..k+3, lane>=16 takes k+4..k+7; the global->local
// K permutation is identical for A and B, so each b128 feeds two WMMAs.
// ---------------------------------------------------------------------------
__global__ __launch_bounds__(128) void sif_gram_wmma_kernel(
    const float* __restrict__ feats,  // [448][32768]
    float* __restrict__ gram)         // [448][448]
{
    __shared__ __align__(32) float red[4][32][8];   // 4 KB partials

    int tid = threadIdx.x;
    int wave = tid >> 5, lane = tid & 31;
    int tile = blockIdx.x;                 // 28*28 = 784 tiles
    int I = tile / 28;
    int J = tile - I * 28;

    int m    = lane & 15;
    int loff = (lane < 16) ? 0 : 4;        // b128 K-permutation offset
    int kbeg = wave << 13;                 // 8192 K per wave

    const float* arow = feats + (((size_t)(I * 16 + m)) << 15) + loff + kbeg;
    const float* brow = feats + (((size_t)(J * 16 + m)) << 15) + loff + kbeg;

    v8f acc = {};
    #pragma unroll 2
    for (int k0 = 0; k0 < 8192; k0 += 16) {
        v4f a0 = *(const v4f*)(arow + k0);
        v4f b0 = *(const v4f*)(brow + k0);
        v4f a1 = *(const v4f*)(arow + k0 + 8);
        v4f b1 = *(const v4f*)(brow + k0 + 8);
        acc = __builtin_amdgcn_wmma_f32_16x16x4_f32(
            false, a0.xy, false, b0.xy, (short)0, acc, false, false);
        acc = __builtin_amdgcn_wmma_f32_16x16x4_f32(
            false, a0.zw, false, b0.zw, (short)0, acc, false, false);
        acc = __builtin_amdgcn_wmma_f32_16x16x4_f32(
            false, a1.xy, false, b1.xy, (short)0, acc, false, false);
        acc = __builtin_amdgcn_wmma_f32_16x16x4_f32(
            false, a1.zw, false, b1.zw, (short)0, acc, false, false);
    }

    *(v8f*)&red[wave][lane][0] = acc;
    __syncthreads();

    if (wave == 0) {
        v8f s = *(const v8f*)&red[0][lane][0];
        s = s + *(const v8f*)&red[1][lane][0];
        s = s + *(const v8f*)&red[2][lane][0];
        s = s + *(const v8f*)&red[3][lane][0];

        const float scale = 1.0f / 14680064.0f;   // 1/(448*32768)
        int roff = (lane < 16) ? 0 : 8;
        #pragma unroll
        for (int v = 0; v < 8; ++v) {
            int M = I * 16 + v + roff;
            int N = J * 16 + m;
            gram[M * GRAM_DIM + N] = s[v] * scale;
        }
    }
}

// ---------------------------------------------------------------------------
extern "C" void kernel_launch(void* const* d_in, const int* in_sizes, int n_in,
                              void* d_out, int out_size, void* d_ws, size_t ws_size,
                              hipStream_t stream) {
    (void)in_sizes; (void)n_in; (void)out_size; (void)d_ws; (void)ws_size;

    const float* image = (const float*)d_in[0];  // [64,1,480,640]
    const float* pup   = (const float*)d_in[1];  // [64,3]
    const float* iri   = (const float*)d_in[2];  // [64,3]
    const float* filt  = (const float*)d_in[3];  // [15,15,7]

    float* out   = (float*)d_out;
    float* codes = out;                                // 14,680,064
    float* gram  = out + CODES_SIZE;                   //    200,704
    float* polar = out + CODES_SIZE + GRAM_SIZE;       //  2,097,152

    // 1) polar bilinear sampling
    sif_polar_kernel<<<POLAR_SIZE / 256, 256, 0, stream>>>(image, pup, iri, polar);

    // 2) wrap conv via async-LDS staging + WMMA: (b,h,wseg) blocks of 4 waves
    sif_conv_wmma_kernel<<<BB * PHH * (PWW / 64), 128, 0, stream>>>(polar, filt, codes);

    // 3) Gram via WMMA: one block per 16x16 tile, K split across 4 waves
    sif_gram_wmma_kernel<<<784, 128, 0, stream>>>(codes, gram);
}